// LearnableSO4_76398878261917
// MI455X (gfx1250) — compile-verified
//
#include <hip/hip_runtime.h>

// Single 4x4 matrix exponential of S = A - A^T (skew-symmetric -> SO(4)).
// One wave32; scaling-and-squaring (fixed 2^-8), 8-term Taylor, 8 squarings.
// Each 4x4 matmul = one zero-padded v_wmma_f32_16x16x4_f32 (K=4 exact match).
// Zero padding lives as data in zeroed stride-32 LDS tiles, so operand loads
// AND result stores are unconditional: lanes 4-31 of the D tile hold exactly
// the zeros the padding needs (B cols >= 4 and A rows >= 4 are zero), and
// with stride 32 their store addresses land in never-read scratch columns.
// Per matmul: 3 ds_load -> wmma -> 2 ds_store_2addr, zero EXEC manipulation.

typedef __attribute__((ext_vector_type(2))) float v2f;
typedef __attribute__((ext_vector_type(4))) float v4f;
typedef __attribute__((ext_vector_type(8))) float v8f;

#define NSCALE 8     // scale S by 2^-8, square result 8 times
#define NTERMS 8     // Taylor terms: I + S + ... + S^8/8!
#define LDSTRIDE 32  // tile row stride (floats); cols 16-31 = store scratch
#define TILEF 512    // floats per padded tile (16 rows x 32 cols)

// A-operand (16x4 f32, 2 VGPRs): lanes 0-15 -> row=lane, K=0,1 ;
// lanes 16-31 -> row=lane-16, K=2,3. Rows >= 4 read pre-zeroed LDS.
__device__ __forceinline__ v2f lds_load_aop(const float* M, int lane) {
  const int row   = lane & 15;
  const int kbase = (lane >> 4) << 1;   // 0 or 2
  v2f a;
  a.x = M[row * LDSTRIDE + kbase + 0];  // consecutive -> ds_load_b64
  a.y = M[row * LDSTRIDE + kbase + 1];
  return a;
}

// B-operand (4x16 f32, 2 VGPRs): lanes 0-15 -> col=lane, rows K=0,1 ;
// lanes 16-31 -> rows K=2,3. Cols >= 4 read pre-zeroed LDS.
__device__ __forceinline__ v2f lds_load_bop(const float* M, int lane) {
  const int col   = lane & 15;
  const int kbase = (lane >> 4) << 1;   // 0 or 2
  v2f b;
  b.x = M[(kbase + 0) * LDSTRIDE + col];
  b.y = M[(kbase + 1) * LDSTRIDE + col];
  return b;
}

// D = X * Y (4x4 live in lanes 0-3, c[0..3]) via one v_wmma_f32_16x16x4_f32.
__device__ __forceinline__ v8f mm4_wmma(const float* X, const float* Y,
                                        int lane) {
  v2f a = lds_load_aop(X, lane);
  v2f b = lds_load_bop(Y, lane);
  v8f c = {};
  // 8 args: (neg_a, A, neg_b, B, c_mod, C, reuse_a, reuse_b)
  return __builtin_amdgcn_wmma_f32_16x16x4_f32(
      false, a, false, b, (short)0, c, false, false);
}

// Unguarded full-wave store of rows 0-3: lane t writes column t.
// Lanes 0-3 write the live 4x4; lanes 4-15 write zeros into the padding
// (refreshing it); lanes 16-31 write into never-read scratch cols 16-31.
__device__ __forceinline__ void store_rows4(float* dst, float r0, float r1,
                                            float r2, float r3, int lane) {
  dst[0 * LDSTRIDE + lane] = r0;
  dst[1 * LDSTRIDE + lane] = r1;
  dst[2 * LDSTRIDE + lane] = r2;
  dst[3 * LDSTRIDE + lane] = r3;
  __syncthreads();  // single-wave workgroup: lowers to waits; barrier is a NOP
}

__launch_bounds__(32)
__global__ void so4_expm_wmma_kernel(const float* __restrict__ A,
                                     float* __restrict__ out) {
  __shared__ float S[TILEF];  // scaled generator S' = (A - A^T) * 2^-NSCALE
  __shared__ float P[TILEF];  // running power term S'^k / k!
  __shared__ float E[TILEF];  // exp(S') tile for the squaring phase
  const int t = (int)threadIdx.x;

  // Zero the padded tiles: 12 x ds_store_b128 per lane (one-time).
  {
    const v4f z = {0.0f, 0.0f, 0.0f, 0.0f};
#pragma unroll
    for (int i = 0; i < TILEF / 128; ++i) {
      ((v4f*)S)[t + 32 * i] = z;
      ((v4f*)P)[t + 32 * i] = z;
      ((v4f*)E)[t + 32 * i] = z;
    }
  }
  __syncthreads();

  if (t < 16) {
    const int r = t >> 2, c4 = t & 3;
    const float s = (A[r * 4 + c4] - A[c4 * 4 + r]) * (1.0f / 256.0f);
    S[r * LDSTRIDE + c4] = s;
    P[r * LDSTRIDE + c4] = s;   // k = 1 term
  }
  __syncthreads();

  // E accumulator lives in registers, lane t = column t.
  // e_m = (I + S')[m][t]; lanes >= 4 read zeros -> stay zero (live padding).
  float e0 = S[0 * LDSTRIDE + t] + ((t == 0) ? 1.0f : 0.0f);
  float e1 = S[1 * LDSTRIDE + t] + ((t == 1) ? 1.0f : 0.0f);
  float e2 = S[2 * LDSTRIDE + t] + ((t == 2) ? 1.0f : 0.0f);
  float e3 = S[3 * LDSTRIDE + t] + ((t == 3) ? 1.0f : 0.0f);
  __syncthreads();

  // Taylor: E = I + S' + S'^2/2! + ... + S'^NTERMS/NTERMS!
#pragma unroll
  for (int k = 2; k <= NTERMS; ++k) {
    v8f c = mm4_wmma(P, S, t);               // c = P_prev * S'
    const float invk = 1.0f / (float)k;
    const float p0 = c[0] * invk;            // fold 1/k -> P = S'^k / k!
    const float p1 = c[1] * invk;
    const float p2 = c[2] * invk;
    const float p3 = c[3] * invk;
    e0 += p0; e1 += p1; e2 += p2; e3 += p3;  // register accumulation
    store_rows4(P, p0, p1, p2, p3, t);       // unguarded, self-padding
  }

  // E registers -> LDS tile once, for the squaring phase.
  store_rows4(E, e0, e1, e2, e3, t);

  // Squaring: exp(S) = (exp(S * 2^-8))^(2^8)
  v8f c;
#pragma unroll
  for (int i = 0; i < NSCALE; ++i) {
    c = mm4_wmma(E, E, t);
    store_rows4(E, c[0], c[1], c[2], c[3], t);
  }

  // Final result is still live in registers on lanes 0-3: straight to global.
  if (t < 4) {
    out[0 * 4 + t] = c[0];
    out[1 * 4 + t] = c[1];
    out[2 * 4 + t] = c[2];
    out[3 * 4 + t] = c[3];
  }
}

extern "C" void kernel_launch(void* const* d_in, const int* in_sizes, int n_in,
                              void* d_out, int out_size, void* d_ws, size_t ws_size,
                              hipStream_t stream) {
  (void)in_sizes; (void)n_in; (void)out_size; (void)d_ws; (void)ws_size;
  const float* A = (const float*)d_in[0];
  float* out = (float*)d_out;
  so4_expm_wmma_kernel<<<1, 32, 0, stream>>>(A, out);
}